// PTS_Loss_52570399703231
// MI455X (gfx1250) — compile-verified
//
#include <hip/hip_runtime.h>
#include <math.h>

typedef __attribute__((ext_vector_type(2))) float v2f;
typedef __attribute__((ext_vector_type(8))) float v8f;

#define NPTS   2048
#define NBATCH 32
#define NTILES (NPTS / 16)      // 128 y-tiles of 16 points
#define SLICES 16               // x-tile slices per batch -> 512 blocks
#define NPART  (NBATCH * SLICES)

__device__ __forceinline__ void quat2mat(const float* __restrict__ q, float* R) {
    float w = q[0], x = q[1], y = q[2], z = q[3];
    float inv = rsqrtf(w * w + x * x + y * y + z * z);
    w *= inv; x *= inv; y *= inv; z *= inv;
    R[0] = 1.f - 2.f * (y * y + z * z); R[1] = 2.f * (x * y - w * z); R[2] = 2.f * (x * z + w * y);
    R[3] = 2.f * (x * y + w * z); R[4] = 1.f - 2.f * (x * x + z * z); R[5] = 2.f * (y * z - w * x);
    R[6] = 2.f * (x * z - w * y); R[7] = 2.f * (y * z + w * x); R[8] = 1.f - 2.f * (x * x + y * y);
}

// grid = (SLICES, NBATCH), block = 256 (8 wave32).
// Each wave owns one 16-point x-tile: tx = slice*8 + wave.
__global__ __launch_bounds__(256) void pts_loss_main(
    const float* __restrict__ q_est, const float* __restrict__ q_gt,
    const float* __restrict__ pts, const int* __restrict__ symmetries,
    float* __restrict__ partial)
{
    __shared__ float4 ypack[NPTS];   // {y0, y1, y2, yy} per gt-rotated point (32 KB)
    __shared__ float  red[8];

    const int slice = blockIdx.x;
    const int b     = blockIdx.y;
    const int tid   = threadIdx.x;
    const int lane  = tid & 31;
    const int wave  = tid >> 5;

    float Re[9], Rg[9];
    quat2mat(q_est + 4 * b, Re);
    quat2mat(q_gt  + 4 * b, Rg);
    const float* __restrict__ P = pts + (size_t)b * NPTS * 3;

    float acc = 0.0f;

    if (symmetries[b] == 0) {
        // -------- ADD path (cheap): only slice 0 computes; others contribute 0 --------
        if (slice == 0) {
            float D0 = Re[0] - Rg[0], D1 = Re[1] - Rg[1], D2 = Re[2] - Rg[2];
            float D3 = Re[3] - Rg[3], D4 = Re[4] - Rg[4], D5 = Re[5] - Rg[5];
            float D6 = Re[6] - Rg[6], D7 = Re[7] - Rg[7], D8 = Re[8] - Rg[8];
            for (int n = tid; n < NPTS; n += 256) {
                float p0 = P[3 * n + 0], p1 = P[3 * n + 1], p2 = P[3 * n + 2];
                float v0 = D0 * p0 + D1 * p1 + D2 * p2;
                float v1 = D3 * p0 + D4 * p1 + D5 * p2;
                float v2 = D6 * p0 + D7 * p1 + D8 * p2;
                acc += v0 * v0 + v1 * v1 + v2 * v2;
            }
        }
    } else {
        // -------- ADD-S path: WMMA pairwise distances + per-column min --------
        // Stage y = Rg*p (+ yy) into LDS, A-fragment friendly: float4 per point.
        for (int n = tid; n < NPTS; n += 256) {
            float p0 = P[3 * n + 0], p1 = P[3 * n + 1], p2 = P[3 * n + 2];
            float y0 = Rg[0] * p0 + Rg[1] * p1 + Rg[2] * p2;
            float y1 = Rg[3] * p0 + Rg[4] * p1 + Rg[5] * p2;
            float y2 = Rg[6] * p0 + Rg[7] * p1 + Rg[8] * p2;
            ypack[n] = float4{y0, y1, y2, y0 * y0 + y1 * y1 + y2 * y2};
        }
        __syncthreads();

        const int hi  = lane >> 4;    // half-wave: selects K-slot pair (0,1) vs (2,3)
        const int col = lane & 15;    // column n within this wave's 16-wide x-tile
        // A fragment: 2 consecutive floats of point m's float4.
        // float offset = m*4 + hi*2, m = ty*16 + col  ->  ty*64 + (col*4 + hi*2)
        const float* yb = reinterpret_cast<const float*>(ypack) + col * 4 + hi * 2;

        const int tx = slice * 8 + wave;      // this wave's x-tile (0..127)
        const int n  = tx * 16 + col;         // lanes L and L+16 load the same point
        float p0 = P[3 * n + 0], p1 = P[3 * n + 1], p2 = P[3 * n + 2];
        float x0 = Re[0] * p0 + Re[1] * p1 + Re[2] * p2;
        float x1 = Re[3] * p0 + Re[4] * p1 + Re[5] * p2;
        float x2 = Re[6] * p0 + Re[7] * p1 + Re[8] * p2;
        float xx = x0 * x0 + x1 * x1 + x2 * x2;

        // B (4x16): column n holds [-2x0; -2x1; -2x2; 1]
        v2f bfrag;
        bfrag.x = hi ? (-2.0f * x2) : (-2.0f * x0);
        bfrag.y = hi ? 1.0f         : (-2.0f * x1);

        v8f run;
        #pragma unroll
        for (int i = 0; i < 8; ++i) run[i] = 3.0e38f;
        v8f cz = {};   // zero accumulator (D = A*B)

        // Double-buffered sweep over all y tiles: prefetch ty+2/ty+3 while
        // the matrix pipe + min3 chain works on ty/ty+1, so s_wait_dscnt for
        // the next fragments lands behind ~16 independent VALU ops.
        v2f a0 = *reinterpret_cast<const v2f*>(yb + 0 * 64);
        v2f a1 = *reinterpret_cast<const v2f*>(yb + 1 * 64);
        for (int ty = 0; ty < NTILES - 2; ty += 2) {
            v2f n0 = *reinterpret_cast<const v2f*>(yb + (ty + 2) * 64);
            v2f n1 = *reinterpret_cast<const v2f*>(yb + (ty + 3) * 64);
            v8f d0 = __builtin_amdgcn_wmma_f32_16x16x4_f32(
                false, a0, false, bfrag, (short)0, cz, false, false);
            v8f d1 = __builtin_amdgcn_wmma_f32_16x16x4_f32(
                false, a1, false, bfrag, (short)0, cz, false, false);
            #pragma unroll
            for (int i = 0; i < 8; ++i)
                run[i] = fminf(run[i], fminf(d0[i], d1[i]));
            a0 = n0;
            a1 = n1;
        }
        {   // epilogue: last two tiles
            v8f d0 = __builtin_amdgcn_wmma_f32_16x16x4_f32(
                false, a0, false, bfrag, (short)0, cz, false, false);
            v8f d1 = __builtin_amdgcn_wmma_f32_16x16x4_f32(
                false, a1, false, bfrag, (short)0, cz, false, false);
            #pragma unroll
            for (int i = 0; i < 8; ++i)
                run[i] = fminf(run[i], fminf(d0[i], d1[i]));
        }

        // min over the 8 m-rows in this lane, then across half-waves
        float t = run[0];
        #pragma unroll
        for (int i = 1; i < 8; ++i) t = fminf(t, run[i]);
        t = fminf(t, __shfl_xor(t, 16, 32));
        float d_s = t + xx;                   // min_m ||x_n - y_m||^2
        acc += (hi == 0) ? d_s : 0.0f;        // count each column once
    }

    // deterministic block reduction; EVERY block writes its partial slot
    #pragma unroll
    for (int off = 16; off >= 1; off >>= 1) acc += __shfl_xor(acc, off, 32);
    if (lane == 0) red[wave] = acc;
    __syncthreads();
    if (tid == 0) {
        float s = 0.0f;
        #pragma unroll
        for (int i = 0; i < 8; ++i) s += red[i];
        partial[b * SLICES + slice] = s;
    }
}

__global__ void pts_loss_finalize(const float* __restrict__ partial, float* __restrict__ out) {
    if (threadIdx.x == 0 && blockIdx.x == 0) {
        float s = 0.0f;
        for (int i = 0; i < NPART; ++i) s += partial[i];
        out[0] = s / (2.0f * (float)NPTS * (float)NBATCH);
    }
}

extern "C" void kernel_launch(void* const* d_in, const int* in_sizes, int n_in,
                              void* d_out, int out_size, void* d_ws, size_t ws_size,
                              hipStream_t stream) {
    const float* q_est = (const float*)d_in[0];
    const float* q_gt  = (const float*)d_in[1];
    // d_in[2] = T : unused by the reference loss
    const float* pts   = (const float*)d_in[3];
    const int*   sym   = (const int*)d_in[4];

    float* partial = (float*)d_ws;   // NPART floats of scratch

    dim3 grid(SLICES, NBATCH);
    pts_loss_main<<<grid, 256, 0, stream>>>(q_est, q_gt, pts, sym, partial);
    pts_loss_finalize<<<1, 32, 0, stream>>>(partial, (float*)d_out);
}